// SimpleGNN_EFG_10557029614292
// MI455X (gfx1250) — compile-verified
//
#include <hip/hip_runtime.h>

// ---------------- CDNA5 WMMA types ----------------
typedef __attribute__((ext_vector_type(2))) float v2f;
typedef __attribute__((ext_vector_type(8))) float v8f;

#define HID 256
#define NGRAPHS 64

__device__ __forceinline__ void atomAddF(float* p, float v) {
  // lowers to global_atomic_add_f32 (no CAS loop)
  unsafeAtomicAdd(p, v);
}

// ---------------- GEMM: Y[M x 256] = act(X)[M x 256] @ W[256 x 256] ----------------
// One wave computes a 16x128 output tile (8 x v8f accumulators), K in steps of 4
// using V_WMMA_F32_16X16X4_F32 (full f32 precision, matches the f32 reference).
// If abias != nullptr the A operand is relu(X + abias[k]) (fuses the previous
// layer's bias+ReLU into this GEMM's A load — saves a full elementwise pass).
// A layout (16x4 MxK): lane l -> row (l&15), VGPR0/1 hold K = (l>>4)*2 + {0,1}.
// B layout (4x16 KxN): lane l -> col (l&15), VGPR0/1 hold K = (l>>4)*2 + {0,1}.
// C/D layout (16x16):  lane l -> col (l&15), VGPR v -> row v + (l>>4)*8.
__global__ void gcn_wmma_gemm(const float* __restrict__ X,
                              const float* __restrict__ W,
                              const float* __restrict__ abias,  // may be nullptr
                              float* __restrict__ Y, int M) {
  const int lane = threadIdx.x & 31;
  const int wave = threadIdx.x >> 5;
  const int waveId = blockIdx.x * (blockDim.x >> 5) + wave;
  const int tilesJ = HID / 128;             // 2 col-groups of 128
  const int tm = waveId / tilesJ;
  const int tj = waveId % tilesJ;
  if (tm * 16 >= M) return;

  const int row   = tm * 16 + (lane & 15);
  const int khalf = (lane >> 4) * 2;        // 0 or 2
  const int col0  = tj * 128 + (lane & 15);

  v8f acc[8] = {};
  for (int k = 0; k < HID; k += 4) {
    v2f a;
    const float* ap = X + (size_t)row * HID + k + khalf;
    a[0] = ap[0];
    a[1] = ap[1];
    if (abias) {  // fused bias + ReLU on the A operand
      float t0 = a[0] + abias[k + khalf];
      float t1 = a[1] + abias[k + khalf + 1];
      a[0] = t0 > 0.0f ? t0 : 0.0f;
      a[1] = t1 > 0.0f ? t1 : 0.0f;
    }
#pragma unroll
    for (int j = 0; j < 8; ++j) {
      v2f b;
      const float* bp = W + (size_t)(k + khalf) * HID + col0 + j * 16;
      b[0] = bp[0];
      b[1] = bp[HID];
      acc[j] = __builtin_amdgcn_wmma_f32_16x16x4_f32(
          /*neg_a=*/false, a, /*neg_b=*/false, b,
          /*c_mod=*/(short)0, acc[j], /*reuse_a=*/false, /*reuse_b=*/false);
    }
  }

  const int mBase = tm * 16 + (lane >> 4) * 8;
#pragma unroll
  for (int j = 0; j < 8; ++j) {
    const int col = tj * 128 + j * 16 + (lane & 15);
#pragma unroll
    for (int v = 0; v < 8; ++v) {
      Y[(size_t)(mBase + v) * HID + col] = acc[j][v];
    }
  }
}

// ---------------- degree / norm ----------------
__global__ void deg_init(float* __restrict__ deg, int N) {
  int i = blockIdx.x * blockDim.x + threadIdx.x;
  if (i < N) deg[i] = 1.0f;  // self-loop
}

__global__ void deg_accum(const int* __restrict__ dst, float* __restrict__ deg, int E) {
  int e = blockIdx.x * blockDim.x + threadIdx.x;
  if (e < E) atomAddF(&deg[dst[e]], 1.0f);
}

__global__ void deg_to_dinv(float* __restrict__ deg, int N) {
  int i = blockIdx.x * blockDim.x + threadIdx.x;
  if (i < N) deg[i] = __frsqrt_rn(fmaxf(deg[i], 1e-12f));
}

// ---------------- zero (float4) ----------------
__global__ void zero_f32v4(float4* __restrict__ p, int count4) {
  int i = blockIdx.x * blockDim.x + threadIdx.x;
  if (i < count4) p[i] = make_float4(0.f, 0.f, 0.f, 0.f);
}

// ---------------- edge scatter: out[dst,:] += dinv[src]*dinv[dst]*h[src,:] ----------------
// One wave per (edge or self-loop). 256 feats = 2 x float4 per lane.
__global__ void edge_scatter(const float* __restrict__ h,
                             const int* __restrict__ src,
                             const int* __restrict__ dst,
                             const float* __restrict__ dinv,
                             float* __restrict__ out, int E, int N) {
  const int lane = threadIdx.x & 31;
  const long long e = (long long)blockIdx.x * (blockDim.x >> 5) + (threadIdx.x >> 5);
  if (e >= (long long)E + N) return;
  int s, d;
  if (e < E) { s = src[e]; d = dst[e]; }
  else       { s = d = (int)(e - E); }
  const float nrm = dinv[s] * dinv[d];
  const float4* hp = (const float4*)(h + (size_t)s * HID);
  float* op = out + (size_t)d * HID;
#pragma unroll
  for (int j = 0; j < 2; ++j) {
    float4 v = hp[lane + j * 32];
    int base = (lane + j * 32) * 4;
    atomAddF(op + base + 0, v.x * nrm);
    atomAddF(op + base + 1, v.y * nrm);
    atomAddF(op + base + 2, v.z * nrm);
    atomAddF(op + base + 3, v.w * nrm);
  }
}

// ---------------- pooling: pooled[batch[n],:] += relu(h[n,:]+bias)  (batch sorted) ----------------
#define NODES_PER_BLOCK 128
__global__ void pool_nodes(const float* __restrict__ h, const int* __restrict__ batch,
                           const float* __restrict__ bias, float* __restrict__ pooled, int N) {
  const int f = threadIdx.x;                 // 256 threads == 256 features
  int n0 = blockIdx.x * NODES_PER_BLOCK;
  if (n0 >= N) return;
  int n1 = n0 + NODES_PER_BLOCK;
  if (n1 > N) n1 = N;
  const float bf = bias[f];
  float acc = 0.0f;
  int cur = batch[n0];
  for (int n = n0; n < n1; ++n) {
    int g = batch[n];
    if (g != cur) {
      atomAddF(&pooled[(size_t)cur * HID + f], acc);
      acc = 0.0f;
      cur = g;
    }
    float v = h[(size_t)n * HID + f] + bf;   // fused bias + ReLU
    acc += (v > 0.0f ? v : 0.0f);
  }
  atomAddF(&pooled[(size_t)cur * HID + f], acc);
}

// ---------------- output head: out[g] = pooled[g,:] . W_out + b_out ----------------
__global__ void out_head(const float* __restrict__ pooled, const float* __restrict__ Wout,
                         const float* __restrict__ bout, float* __restrict__ out) {
  int g = blockIdx.x * blockDim.x + threadIdx.x;
  if (g >= NGRAPHS) return;
  float s = bout[0];
  for (int f = 0; f < HID; ++f) s += pooled[(size_t)g * HID + f] * Wout[f];
  out[g] = s;
}

static inline unsigned cdiv(long long a, long long b) { return (unsigned)((a + b - 1) / b); }

extern "C" void kernel_launch(void* const* d_in, const int* in_sizes, int n_in,
                              void* d_out, int out_size, void* d_ws, size_t ws_size,
                              hipStream_t stream) {
  const float* x     = (const float*)d_in[0];
  const int*   eidx  = (const int*)d_in[1];
  const int*   batch = (const int*)d_in[2];
  const float* W1    = (const float*)d_in[3];
  const float* b1    = (const float*)d_in[4];
  const float* W2    = (const float*)d_in[5];
  const float* b2    = (const float*)d_in[6];
  const float* Wout  = (const float*)d_in[7];
  const float* bout  = (const float*)d_in[8];
  float* out = (float*)d_out;

  const int N = in_sizes[0] / HID;      // 50000
  const int E = in_sizes[1] / 2;        // 800000
  const int* src = eidx;
  const int* dst = eidx + E;

  // workspace layout
  float* bufA   = (float*)d_ws;                       // N*HID
  float* bufB   = bufA + (size_t)N * HID;             // N*HID
  float* dinv   = bufB + (size_t)N * HID;             // N (deg -> dinv in place)
  float* pooled = dinv + N;                           // NGRAPHS*HID

  const int feat = N * HID;
  const long long totEdges = (long long)E + N;
  const long long gemmWaves = (long long)(N / 16) * (HID / 128);  // 16x128 tile/wave

  // ---- degrees & D^{-1/2} ----
  deg_init<<<cdiv(N, 256), 256, 0, stream>>>(dinv, N);
  deg_accum<<<cdiv(E, 256), 256, 0, stream>>>(dst, dinv, E);
  deg_to_dinv<<<cdiv(N, 256), 256, 0, stream>>>(dinv, N);

  // ---- layer 1: h1 = x @ W1 ; agg1 = A_hat h1 ----
  gcn_wmma_gemm<<<cdiv(gemmWaves, 8), 256, 0, stream>>>(x, W1, nullptr, bufA, N);
  zero_f32v4<<<cdiv(feat / 4, 256), 256, 0, stream>>>((float4*)bufB, feat / 4);
  edge_scatter<<<cdiv(totEdges, 8), 256, 0, stream>>>(bufA, src, dst, dinv, bufB, E, N);

  // ---- layer 2: h2 = relu(agg1 + b1) @ W2 ; agg2 = A_hat h2 ----
  gcn_wmma_gemm<<<cdiv(gemmWaves, 8), 256, 0, stream>>>(bufB, W2, b1, bufA, N);
  zero_f32v4<<<cdiv(feat / 4, 256), 256, 0, stream>>>((float4*)bufB, feat / 4);
  edge_scatter<<<cdiv(totEdges, 8), 256, 0, stream>>>(bufA, src, dst, dinv, bufB, E, N);

  // ---- pooling (fused bias2 + ReLU) + head ----
  zero_f32v4<<<cdiv(NGRAPHS * HID / 4, 256), 256, 0, stream>>>((float4*)pooled, NGRAPHS * HID / 4);
  pool_nodes<<<cdiv(N, NODES_PER_BLOCK), HID, 0, stream>>>(bufB, batch, b2, pooled, N);
  out_head<<<1, 64, 0, stream>>>(pooled, Wout, bout, out);
}